// InputAttentionLayer_38139309588726
// MI455X (gfx1250) — compile-verified
//
#include <hip/hip_runtime.h>

// Shapes: P=2,S=2,C=64,F=64,B=64 ; PS = P*S = 4
typedef float v2f __attribute__((ext_vector_type(2)));
typedef float v4f __attribute__((ext_vector_type(4)));
typedef float v8f __attribute__((ext_vector_type(8)));

// ---------------------------------------------------------------------------
// Kernel 1: scores[ps,c,d] = sum_{f1,f2} dbfc[ps,c,f1,d,f2], then softmax over d.
// One block per (ps*64 + c) slab of 64*64*64 floats (1 MB). 1024 threads:
// tid = d*16 + j ; thread streams column-chunk j (4 floats) of every f1 row
// for its d, with non-temporal float4 loads (one-touch 256 MB stream,
// ~11.5 us at 23.3 TB/s == the whole-problem roofline).
// ---------------------------------------------------------------------------
__global__ __launch_bounds__(1024) void k_scores_softmax(
    const float* __restrict__ dbfc, float* __restrict__ weights) {
  __shared__ float part[1024];
  __shared__ float sc[64];
  __shared__ float ex[64];

  const int tid = threadIdx.x;
  const int d = tid >> 4;   // 0..63
  const int j = tid & 15;   // float4 chunk within the 64-float f2 row
  const float* base =
      dbfc + (size_t)blockIdx.x * (64 * 64 * 64) + d * 64 + j * 4;

  float acc = 0.f;
#pragma unroll 8
  for (int f1 = 0; f1 < 64; ++f1) {
    v4f x = __builtin_nontemporal_load((const v4f*)(base + f1 * 4096));
    acc += x.x + x.y + x.z + x.w;
  }
  part[tid] = acc;
  __syncthreads();

  if (tid < 64) {
    float s = 0.f;
#pragma unroll
    for (int q = 0; q < 16; ++q) s += part[tid * 16 + q];
    sc[tid] = s;
  }
  __syncthreads();
  if (tid < 64) {
    float m = -1e30f;
    for (int q = 0; q < 64; ++q) m = fmaxf(m, sc[q]);
    ex[tid] = __expf(sc[tid] - m);
  }
  __syncthreads();
  if (tid < 64) {
    float s = 0.f;
    for (int q = 0; q < 64; ++q) s += ex[q];
    weights[blockIdx.x * 64 + tid] = ex[tid] / s;  // weights[ps,c,d]
  }
}

// ---------------------------------------------------------------------------
// Kernel 2: per (b, ps):  C1[c,k] = sum_d W[ps,c,d] * X[b,ps,d,k]
//                         Out[c,f] = sum_k C1[c,k] * wV[f,k]
// Both 64x64x64 GEMMs via V_WMMA_F32_16X16X4_F32 (fp32, K=4/step, 16 steps).
// 256 threads = 8 waves; wave w owns tiles w and w+8 (same n0, m0 +/- 32):
// the two tiles share every B fragment and give two independent WMMA
// accumulator chains per wave (2x ILP, no exec-masked tile loop).
// LDS pitch 65 breaks the same-bank stride-64 column reads.
// Output: out[c, b, ps*64+f] with row length B*PS*F = 16384.
// ---------------------------------------------------------------------------
__global__ __launch_bounds__(256) void k_attn_gemm(
    const float* __restrict__ X, const float* __restrict__ wV,
    const float* __restrict__ weights, float* __restrict__ out) {
  const int b = blockIdx.x >> 2;
  const int ps = blockIdx.x & 3;
  const int tid = threadIdx.x;
  const int lane = tid & 31;          // wave32
  const int w = tid >> 5;             // 8 waves
  const int half = lane >> 4;         // 0/1 : K-pair select for A/B fragments
  const int l15 = lane & 15;

  __shared__ float Wl[64 * 65];   // softmax weights for this ps
  __shared__ float Xl[64 * 65];   // X[b, ps] as [d][k]
  __shared__ float Vl[64 * 65];   // wV as [f][k]
  __shared__ float C1[64 * 65];   // intermediate [c][k]

  const float* Wg = weights + ps * 4096;
  const float* Xg = X + (size_t)b * 16384 + ps * 4096;
#pragma unroll
  for (int e = tid; e < 4096; e += 256) {
    const int r = e >> 6, cc = e & 63;
    Wl[r * 65 + cc] = Wg[e];
    Xl[r * 65 + cc] = Xg[e];
    Vl[r * 65 + cc] = wV[e];
  }
  __syncthreads();

  // Tile geometry: tiles t0 = w, t1 = w + 8  ->  same n0, m1 = m0 + 32
  const int m0 = (w >> 2) * 16;       // 0 or 16
  const int m1 = m0 + 32;             // 32 or 48
  const int n0 = (w & 3) * 16;
  const int arow0 = (m0 + l15) * 65;
  const int arow1 = (m1 + l15) * 65;
  const int bcol = n0 + l15;
  const int crow0 = (m0 + half * 8) * 65 + bcol;  // C/D store base, tile 0
  const int crow1 = (m1 + half * 8) * 65 + bcol;  // C/D store base, tile 1

  // Stage 1: C1 = W @ Xb
  {
    v8f acc0 = {}, acc1 = {};
#pragma unroll
    for (int kk = 0; kk < 16; ++kk) {
      const int kb = kk * 4 + half * 2;
      v2f bb = {Xl[kb * 65 + bcol], Xl[(kb + 1) * 65 + bcol]};
      v2f a0 = {Wl[arow0 + kb], Wl[arow0 + kb + 1]};
      v2f a1 = {Wl[arow1 + kb], Wl[arow1 + kb + 1]};
      acc0 = __builtin_amdgcn_wmma_f32_16x16x4_f32(false, a0, false, bb,
                                                   (short)0, acc0, false, false);
      acc1 = __builtin_amdgcn_wmma_f32_16x16x4_f32(false, a1, false, bb,
                                                   (short)0, acc1, false, false);
    }
#pragma unroll
    for (int i = 0; i < 8; ++i) {
      C1[crow0 + i * 65] = acc0[i];
      C1[crow1 + i * 65] = acc1[i];
    }
  }
  __syncthreads();

  // Stage 2: Out = C1 @ wV^T   (B[k,f] = wV[f,k], shared between both tiles)
  {
    v8f acc0 = {}, acc1 = {};
#pragma unroll
    for (int kk = 0; kk < 16; ++kk) {
      const int kb = kk * 4 + half * 2;
      v2f bb = {Vl[bcol * 65 + kb], Vl[bcol * 65 + kb + 1]};
      v2f a0 = {C1[arow0 + kb], C1[arow0 + kb + 1]};
      v2f a1 = {C1[arow1 + kb], C1[arow1 + kb + 1]};
      acc0 = __builtin_amdgcn_wmma_f32_16x16x4_f32(false, a0, false, bb,
                                                   (short)0, acc0, false, false);
      acc1 = __builtin_amdgcn_wmma_f32_16x16x4_f32(false, a1, false, bb,
                                                   (short)0, acc1, false, false);
    }
    const size_t obase = (size_t)b * 256 + ps * 64 + bcol;
#pragma unroll
    for (int i = 0; i < 8; ++i) {
      const int c0 = m0 + half * 8 + i;
      const int c1 = m1 + half * 8 + i;
      out[(size_t)c0 * 16384 + obase] = acc0[i];
      out[(size_t)c1 * 16384 + obase] = acc1[i];
    }
  }
}

extern "C" void kernel_launch(void* const* d_in, const int* in_sizes, int n_in,
                              void* d_out, int out_size, void* d_ws,
                              size_t ws_size, hipStream_t stream) {
  (void)in_sizes; (void)n_in; (void)out_size; (void)ws_size;
  const float* X = (const float*)d_in[0];
  const float* dbfc = (const float*)d_in[1];
  const float* wV = (const float*)d_in[2];
  float* out = (float*)d_out;
  float* weights = (float*)d_ws;  // 4*64*64 floats = 64 KB scratch

  // Bandwidth-bound reduction + fused softmax: 256 blocks x 1024 threads
  k_scores_softmax<<<256, 1024, 0, stream>>>(dbfc, weights);
  // Fused double-GEMM attention apply: 256 blocks (b, ps) x 256 threads
  k_attn_gemm<<<256, 256, 0, stream>>>(X, wV, weights, out);
}